// PoolWithHole_43971875176514
// MI455X (gfx1250) — compile-verified
//
#include <hip/hip_runtime.h>
#include <cstddef>

// ---------------------------------------------------------------------------
// PoolWithHole: out = max(0, max of 8 neighbors (3x3 minus center)), zero pad.
// Shape: B=64, H=1024, W=1024, fp32. Purely HBM-bandwidth bound (~22us floor
// at 23.3 TB/s). Strategy: LDS-staged row tiles, 128-bit transactions, and
// the gfx1250 async global->LDS copy path (ASYNCcnt) for staging.
// ---------------------------------------------------------------------------

#define POOL_W 1024
#define POOL_H 1024
#define TILE_H 8                 // output rows per workgroup
#define LDS_ROWS (TILE_H + 2)    // staged rows incl. top/bottom halo
// Row stride in floats: interior starts at +4 floats (16B) so every b128 LDS
// access is 16B aligned. 1036*4 bytes is a multiple of 16; 1036 % 64 = 12
// banks of skew per row -> no systematic bank conflicts.
#define LDS_STRIDE 1036
#define LDS_LEFT_HALO 3          // zeroed column (col -1)
#define LDS_INTERIOR 4           // interior cols live at [4 .. 4+W-1]

// --- gfx1250 async global->LDS copy (GLOBAL_LOAD_ASYNC_TO_LDS_B128) --------
// Builtin signature (from hipcc diagnostic): first param is
// 'int __vector(4) addrspace(1)*' (global, 128-bit granule). Pass matching
// vector-typed pointers with explicit address spaces.
#if defined(__gfx1250__) && __has_builtin(__builtin_amdgcn_global_load_async_to_lds_b128)
#define POOL_USE_ASYNC 1
typedef int pool_v4i __attribute__((ext_vector_type(4)));
typedef __attribute__((address_space(1))) pool_v4i pool_gv4i;  // global 128-bit
typedef __attribute__((address_space(3))) pool_v4i pool_lv4i;  // LDS 128-bit
#else
#define POOL_USE_ASYNC 0
#endif

__global__ __launch_bounds__(256) void PoolWithHole_43971875176514_kernel(
    const float* __restrict__ x, float* __restrict__ out, int batches) {
  __shared__ float tile[LDS_ROWS * LDS_STRIDE];  // ~41.4 KB

  const int tid  = threadIdx.x;            // 0..255, one float4 column group
  const int row0 = blockIdx.x * TILE_H;    // first output row of this tile
  const int b    = blockIdx.y;             // batch image
  const int c    = tid * 4;                // first column of this lane's group

  const float* __restrict__ xb = x + (size_t)b * POOL_H * POOL_W;

  // Zero the halo columns (col -1 and col W) for all staged rows so the
  // zero-padded border falls out of the max for free.
  if (tid < LDS_ROWS) {
    tile[tid * LDS_STRIDE + LDS_LEFT_HALO]          = 0.0f;
    tile[tid * LDS_STRIDE + LDS_INTERIOR + POOL_W]  = 0.0f;
  }

  // ---- Stage (TILE_H + 2) rows of the image into LDS -----------------------
  // One b128 per lane per row. In-range rows go through the async DMA path
  // (no VGPR round trip, tracked by ASYNCcnt); out-of-range rows (above row 0
  // / below row H-1) are zero-filled with a normal ds store.
#pragma unroll
  for (int lr = 0; lr < LDS_ROWS; ++lr) {
    const int gr = row0 - 1 + lr;
    float* dst = &tile[lr * LDS_STRIDE + LDS_INTERIOR + c];
    if (gr >= 0 && gr < POOL_H) {
      const float* src = xb + (size_t)gr * POOL_W + c;
#if POOL_USE_ASYNC
      __builtin_amdgcn_global_load_async_to_lds_b128(
          (pool_gv4i*)src, (pool_lv4i*)dst, /*offset=*/0, /*cpol=*/0);
#else
      *(float4*)dst = *(const float4*)src;
#endif
    } else {
      float4 z = {0.0f, 0.0f, 0.0f, 0.0f};
      *(float4*)dst = z;
    }
  }

#if POOL_USE_ASYNC
  // Each wave drains its own ASYNCcnt, then the workgroup barrier makes every
  // wave's staged data visible to all.
#if __has_builtin(__builtin_amdgcn_s_wait_asynccnt)
  __builtin_amdgcn_s_wait_asynccnt(0);
#else
  asm volatile("s_wait_asynccnt 0" ::: "memory");
#endif
#endif
  __syncthreads();

  // ---- Compute: 8 output rows per thread, one float4 per row ---------------
  float* __restrict__ ob = out + ((size_t)b * POOL_H + row0) * POOL_W + c;

#pragma unroll
  for (int tr = 0; tr < TILE_H; ++tr) {
    // Windows cover columns c-1 .. c+4 (index 0 == col c-1).
    const float* top = &tile[(tr + 0) * LDS_STRIDE + LDS_LEFT_HALO + c];
    const float* mid = &tile[(tr + 1) * LDS_STRIDE + LDS_LEFT_HALO + c];
    const float* bot = &tile[(tr + 2) * LDS_STRIDE + LDS_LEFT_HALO + c];

    const float t0 = top[0], t1 = top[1], t2 = top[2],
                t3 = top[3], t4 = top[4], t5 = top[5];
    const float b0 = bot[0], b1 = bot[1], b2 = bot[2],
                b3 = bot[3], b4 = bot[4], b5 = bot[5];
    const float m0 = mid[0], m1 = mid[1], m2 = mid[2],
                m3 = mid[3], m4 = mid[4], m5 = mid[5];

    float4 r;
    // out[j] = max(top[j..j+2], bot[j..j+2], mid[j], mid[j+2], 0)
    r.x = fmaxf(fmaxf(fmaxf(t0, t1), fmaxf(t2, b0)),
                fmaxf(fmaxf(b1, b2), fmaxf(m0, m2)));
    r.y = fmaxf(fmaxf(fmaxf(t1, t2), fmaxf(t3, b1)),
                fmaxf(fmaxf(b2, b3), fmaxf(m1, m3)));
    r.z = fmaxf(fmaxf(fmaxf(t2, t3), fmaxf(t4, b2)),
                fmaxf(fmaxf(b3, b4), fmaxf(m2, m4)));
    r.w = fmaxf(fmaxf(fmaxf(t3, t4), fmaxf(t5, b3)),
                fmaxf(fmaxf(b4, b5), fmaxf(m3, m5)));
    r.x = fmaxf(r.x, 0.0f);
    r.y = fmaxf(r.y, 0.0f);
    r.z = fmaxf(r.z, 0.0f);
    r.w = fmaxf(r.w, 0.0f);

    *(float4*)(ob + (size_t)tr * POOL_W) = r;
  }
}

extern "C" void kernel_launch(void* const* d_in, const int* in_sizes, int n_in,
                              void* d_out, int out_size, void* d_ws, size_t ws_size,
                              hipStream_t stream) {
  (void)n_in; (void)out_size; (void)d_ws; (void)ws_size;
  const float* x = (const float*)d_in[0];
  float* out = (float*)d_out;
  const int batches = in_sizes[0] / (POOL_H * POOL_W);  // 64

  dim3 grid(POOL_H / TILE_H, batches, 1);   // 128 x 64 workgroups
  dim3 block(256, 1, 1);                    // 8 wave32s
  PoolWithHole_43971875176514_kernel<<<grid, block, 0, stream>>>(x, out, batches);
}